// CLRerHead_84430467105168
// MI455X (gfx1250) — compile-verified
//
#include <hip/hip_runtime.h>
#include <math.h>

#define B_      32
#define C_      64
#define NP_     192
#define NO_     72
#define HD_     64
#define REGD_   76
#define OUTD_   79
#define SA_     68   // LDS row stride (floats): (68*m+k)%64 avoids frag-read bank conflicts

typedef float v2f __attribute__((ext_vector_type(2)));
typedef float v8f __attribute__((ext_vector_type(8)));

// CDNA5 fp32 WMMA: D(16x16,f32) = A(16x4,f32) * B(4x16,f32) + C
// args: (neg_a, A, neg_b, B, c_mod, C, reuse_a, reuse_b)
__device__ __forceinline__ v8f wmma4(v2f a, v2f b, v8f c) {
  return __builtin_amdgcn_wmma_f32_16x16x4_f32(false, a, false, b, (short)0, c, false, false);
}

// ---------------------------------------------------------------------------
// Kernel A: bilinear sample (72x64) -> gate -> GEMM 64x64 (WMMA f32) -> relu
//           -> pairwise max pool -> mean over 36 -> accumulate fusedSum
// one workgroup per (b,n); 4 waves, each owns a 16-column N-tile
// ---------------------------------------------------------------------------
__global__ void __launch_bounds__(128)
stage_sample_gemm(const float* __restrict__ feat, int H, int W,
                  const float* __restrict__ prior, int priorPerB,
                  const float* __restrict__ l_weight,
                  const float* __restrict__ lsgi,      // [64,64] for this stage
                  float* __restrict__ fusedSum, int firstStage)
{
  __shared__ float sA[80 * SA_];
  __shared__ float s_wx[NO_], s_wy[NO_], s_gate[NO_];
  __shared__ int   s_x0[NO_], s_x1[NO_], s_y0[NO_], s_y1[NO_];

  const int tid  = threadIdx.x;
  const int lane = tid & 31;
  const int wave = tid >> 5;
  const int bn   = blockIdx.x;
  const int b    = bn / NP_;
  const int n    = bn - b * NP_;

  // per-o sampling coefficients: iy = ys[o]*(H-1), ix = prior_x*(W-1)
  if (tid < NO_) {
    int o = tid;
    float ys  = 1.f - (float)o * (1.f / 71.f);
    float iy  = ys * (float)(H - 1);
    float iyf = floorf(iy);
    int y0 = (int)iyf; y0 = y0 < 0 ? 0 : (y0 > H - 1 ? H - 1 : y0);
    int y1 = (y0 + 1 > H - 1) ? H - 1 : y0 + 1;
    s_wy[o] = iy - iyf; s_y0[o] = y0; s_y1[o] = y1;
    float px  = priorPerB ? prior[(size_t)bn * NO_ + o] : prior[n * NO_ + o];
    float ix  = px * (float)(W - 1);
    float ixf = floorf(ix);
    int x0 = (int)ixf; x0 = x0 < 0 ? 0 : (x0 > W - 1 ? W - 1 : x0);
    int x1 = (x0 + 1 > W - 1) ? W - 1 : x0 + 1;
    s_wx[o] = ix - ixf; s_x0[o] = x0; s_x1[o] = x1;
    s_gate[o] = 1.f / (1.f + expf(-l_weight[o]));
  }
  __syncthreads();

  // cooperative bilinear sampling into A tile (rows = o, cols = c)
  const float* fb = feat + (size_t)b * C_ * H * W;
  for (int idx = tid; idx < C_ * NO_; idx += 128) {
    int c = idx / NO_;
    int o = idx - c * NO_;
    const float* fp = fb + (size_t)c * H * W;
    int y0 = s_y0[o], y1 = s_y1[o], x0 = s_x0[o], x1 = s_x1[o];
    float wx = s_wx[o], wy = s_wy[o];
    float v00 = fp[y0 * W + x0], v01 = fp[y0 * W + x1];
    float v10 = fp[y1 * W + x0], v11 = fp[y1 * W + x1];
    float v = v00 * (1.f - wx) * (1.f - wy) + v01 * wx * (1.f - wy)
            + v10 * (1.f - wx) * wy + v11 * wx * wy;
    sA[o * SA_ + c] = v * s_gate[o];
  }
  // zero-pad rows 72..79 so the last M-tile is harmless
  for (int idx = tid; idx < 8 * C_; idx += 128) {
    int r = 72 + (idx >> 6);
    sA[r * SA_ + (idx & 63)] = 0.f;
  }
  __syncthreads();

  // preload 16 fp32 B-fragments (this wave's 16-column block) into registers
  const int ncol = (wave << 4) + (lane & 15);
  const int koff = (lane >> 4) << 1;                 // 0 (lanes 0-15) / 2 (16-31)
  v2f bfr[16];
#pragma unroll
  for (int k = 0; k < 16; k++) {
    bfr[k].x = lsgi[(4 * k + koff)     * 64 + ncol];
    bfr[k].y = lsgi[(4 * k + koff + 1) * 64 + ncol];
  }

  float sum = 0.f;
  const int arow = lane & 15;
#pragma unroll
  for (int mt = 0; mt < 5; mt++) {
    v8f acc = {};
#pragma unroll
    for (int k = 0; k < 16; k++) {
      v2f a;
      const float* ap = &sA[(mt * 16 + arow) * SA_ + 4 * k + koff];
      a.x = ap[0]; a.y = ap[1];
      acc = wmma4(a, bfr[k], acc);
    }
    // relu + pairwise max along o (adjacent C VGPRs = adjacent rows) + row-sum
    float p0 = fmaxf(fmaxf(acc[0], 0.f), fmaxf(acc[1], 0.f));
    float p1 = fmaxf(fmaxf(acc[2], 0.f), fmaxf(acc[3], 0.f));
    float p2 = fmaxf(fmaxf(acc[4], 0.f), fmaxf(acc[5], 0.f));
    float p3 = fmaxf(fmaxf(acc[6], 0.f), fmaxf(acc[7], 0.f));
    float part = p0 + p1 + p2 + p3;
    if (mt == 4 && lane >= 16) part = 0.f;           // padded rows 72..79
    sum += part;
  }
  sum += __shfl_xor(sum, 16, 32);                    // fold lane halves (wave32)
  float compMean = sum * (1.f / 36.f);
  if (lane < 16) {
    float* fs = fusedSum + (size_t)bn * C_ + ncol;
    if (firstStage) *fs = compMean; else *fs += compMean;
  }
}

// ---------------------------------------------------------------------------
// Kernel B helpers: 64-row LDS->LDS WMMA layer and masked projection
// ---------------------------------------------------------------------------
__device__ __forceinline__ void layer64(const float* lA, const float* __restrict__ gW,
                                        const float* __restrict__ gBias,
                                        float* lOut, int lane, int wave)
{
  const int arow = (wave << 4) + (lane & 15);
  const int koff = (lane >> 4) << 1;
  for (int nt = 0; nt < 4; nt++) {
    const int col = (nt << 4) + (lane & 15);
    v2f bfr[16];
#pragma unroll
    for (int k = 0; k < 16; k++) {
      bfr[k].x = gW[(4 * k + koff)     * 64 + col];
      bfr[k].y = gW[(4 * k + koff + 1) * 64 + col];
    }
    v8f acc = {};
#pragma unroll
    for (int k = 0; k < 16; k++) {
      v2f a;
      const float* ap = &lA[arow * SA_ + 4 * k + koff];
      a.x = ap[0]; a.y = ap[1];
      acc = wmma4(a, bfr[k], acc);
    }
    const float bb = gBias[col];
#pragma unroll
    for (int j = 0; j < 8; j++) {
      int row = (wave << 4) + j + ((lane >> 4) << 3);
      lOut[row * SA_ + col] = fmaxf(acc[j] + bb, 0.f);
    }
  }
}

__device__ __forceinline__ void project(const float* lA, const float* __restrict__ gW,
                                        const float* __restrict__ gBias, int N,
                                        float* lOut, int lane, int wave)
{
  const int arow = (wave << 4) + (lane & 15);
  const int koff = (lane >> 4) << 1;
  const int ntiles = (N + 15) >> 4;
  for (int nt = 0; nt < ntiles; nt++) {
    const int col = (nt << 4) + (lane & 15);
    const bool cv = col < N;
    v2f bfr[16];
#pragma unroll
    for (int k = 0; k < 16; k++) {
      bfr[k].x = cv ? gW[(4 * k + koff)     * N + col] : 0.f;
      bfr[k].y = cv ? gW[(4 * k + koff + 1) * N + col] : 0.f;
    }
    v8f acc = {};
#pragma unroll
    for (int k = 0; k < 16; k++) {
      v2f a;
      const float* ap = &lA[arow * SA_ + 4 * k + koff];
      a.x = ap[0]; a.y = ap[1];
      acc = wmma4(a, bfr[k], acc);
    }
    const float bb = cv ? gBias[col] : 0.f;
#pragma unroll
    for (int j = 0; j < 8; j++) {
      int row = (wave << 4) + j + ((lane >> 4) << 3);
      lOut[row * 80 + col] = acc[j] + bb;
    }
  }
}

// ---------------------------------------------------------------------------
// Kernel B: fused mean -> fc -> {cls,reg,loc} MLPs -> outputs + next priors
// one workgroup per 64 rows of the 6144 (b,n) rows
// ---------------------------------------------------------------------------
__global__ void __launch_bounds__(128)
head_kernel(const float* __restrict__ fusedSum, float invS,
            const float* __restrict__ fc_w,  const float* __restrict__ fc_b,
            const float* __restrict__ cls_w, const float* __restrict__ cls_b,
            const float* __restrict__ cls_ow, const float* __restrict__ cls_ob,
            const float* __restrict__ reg_w, const float* __restrict__ reg_b,
            const float* __restrict__ reg_ow, const float* __restrict__ reg_ob,
            const float* __restrict__ loc_w, const float* __restrict__ loc_b,
            const float* __restrict__ loc_ow, const float* __restrict__ loc_ob,
            float* __restrict__ out,          // [6144,79] for this stage
            float* __restrict__ priorNext)    // [6144,72]
{
  __shared__ float sIn[64 * SA_];
  __shared__ float sFC[64 * SA_];
  __shared__ float sT1[64 * SA_];
  __shared__ float sT2[64 * SA_];
  __shared__ float sOut[64 * 80];

  const int tid  = threadIdx.x;
  const int lane = tid & 31;
  const int wave = tid >> 5;
  const int r0   = blockIdx.x * 64;

  for (int idx = tid; idx < 64 * 64; idx += 128) {
    int r = idx >> 6, c = idx & 63;
    sIn[r * SA_ + c] = fusedSum[(size_t)(r0 + r) * 64 + c] * invS;
  }
  __syncthreads();

  layer64(sIn, fc_w, fc_b, sFC, lane, wave);
  __syncthreads();

  // ---- cls (N=2) ----
  layer64(sFC, cls_w,        cls_b,      sT1, lane, wave); __syncthreads();
  layer64(sT1, cls_w + 4096, cls_b + 64, sT2, lane, wave); __syncthreads();
  project(sT2, cls_ow, cls_ob, 2, sOut, lane, wave);       __syncthreads();
  for (int idx = tid; idx < 64 * 2; idx += 128) {
    int r = idx >> 1, c = idx & 1;
    out[(size_t)(r0 + r) * OUTD_ + c] = sOut[r * 80 + c];
  }
  __syncthreads();

  // ---- reg (N=76) + next-stage priors = sigmoid(reg[4:]) ----
  layer64(sFC, reg_w,        reg_b,      sT1, lane, wave); __syncthreads();
  layer64(sT1, reg_w + 4096, reg_b + 64, sT2, lane, wave); __syncthreads();
  project(sT2, reg_ow, reg_ob, REGD_, sOut, lane, wave);   __syncthreads();
  for (int idx = tid; idx < 64 * REGD_; idx += 128) {
    int r = idx / REGD_, c = idx - r * REGD_;
    float v = sOut[r * 80 + c];
    out[(size_t)(r0 + r) * OUTD_ + 2 + c] = v;
    if (c >= 4)
      priorNext[(size_t)(r0 + r) * NO_ + (c - 4)] = 1.f / (1.f + expf(-v));
  }
  __syncthreads();

  // ---- loc (N=1) ----
  layer64(sFC, loc_w,        loc_b,      sT1, lane, wave); __syncthreads();
  layer64(sT1, loc_w + 4096, loc_b + 64, sT2, lane, wave); __syncthreads();
  project(sT2, loc_ow, loc_ob, 1, sOut, lane, wave);       __syncthreads();
  for (int idx = tid; idx < 64; idx += 128)
    out[(size_t)(r0 + idx) * OUTD_ + 78] = sOut[idx * 80];
}

// ---------------------------------------------------------------------------
extern "C" void kernel_launch(void* const* d_in, const int* in_sizes, int n_in,
                              void* d_out, int out_size, void* d_ws, size_t ws_size,
                              hipStream_t stream)
{
  (void)in_sizes; (void)n_in; (void)out_size; (void)ws_size;
  const float* x0     = (const float*)d_in[0];
  const float* x1     = (const float*)d_in[1];
  const float* x2     = (const float*)d_in[2];
  const float* prior0 = (const float*)d_in[3];
  const float* l_wt   = (const float*)d_in[4];
  const float* lsgi   = (const float*)d_in[5];
  const float* fc_w   = (const float*)d_in[6];
  const float* fc_b   = (const float*)d_in[7];
  const float* cls_w  = (const float*)d_in[8];
  const float* cls_b  = (const float*)d_in[9];
  const float* cls_ow = (const float*)d_in[10];
  const float* cls_ob = (const float*)d_in[11];
  const float* reg_w  = (const float*)d_in[12];
  const float* reg_b  = (const float*)d_in[13];
  const float* reg_ow = (const float*)d_in[14];
  const float* reg_ob = (const float*)d_in[15];
  const float* loc_w  = (const float*)d_in[16];
  const float* loc_b  = (const float*)d_in[17];
  const float* loc_ow = (const float*)d_in[18];
  const float* loc_ob = (const float*)d_in[19];
  float* out = (float*)d_out;

  float* fusedSum = (float*)d_ws;                          // 6144*64 f32
  float* priorWS  = fusedSum + (size_t)B_ * NP_ * C_;      // 6144*72 f32

  const float* feats[3] = { x2, x1, x0 };
  const int Hs[3] = { 10, 20, 40 };
  const int Ws_[3] = { 25, 50, 100 };

  for (int s = 0; s < 3; s++) {
    const float* prior = (s == 0) ? prior0 : priorWS;
    stage_sample_gemm<<<B_ * NP_, 128, 0, stream>>>(
        feats[s], Hs[s], Ws_[s], prior, s == 0 ? 0 : 1,
        l_wt, lsgi + (size_t)s * C_ * C_, fusedSum, s == 0 ? 1 : 0);
    head_kernel<<<(B_ * NP_) / 64, 128, 0, stream>>>(
        fusedSum, 1.f / (float)(s + 1),
        fc_w, fc_b, cls_w, cls_b, cls_ow, cls_ob,
        reg_w, reg_b, reg_ow, reg_ob, loc_w, loc_b, loc_ow, loc_ob,
        out + (size_t)s * B_ * NP_ * OUTD_, priorWS);
  }
}